// Model_11888469475921
// MI455X (gfx1250) — compile-verified
//
#include <hip/hip_runtime.h>
#include <math.h>

// SELU constants
#define SELU_SCALE 1.0507009873554805f
#define SELU_ALPHA 1.6732632423543772f

// Problem constants (from reference)
#define NB 1024     // batch
#define CIN 3
#define HH 32
#define WW 32
#define OUTF 64
#define KDIM 27     // 3*3*3 im2col depth

typedef float v2f __attribute__((ext_vector_type(2)));
typedef float v8f __attribute__((ext_vector_type(8)));

// Compile-time im2col decomposition tables: k -> (channel, dh, dw).
// Index >=27 is a masked dummy slot (K padded to 28). Only ever read in
// constexpr context, so they fold to immediates.
constexpr int KC[28]  = {0,0,0,0,0,0,0,0,0, 1,1,1,1,1,1,1,1,1, 2,2,2,2,2,2,2,2,2, 0};
constexpr int KDH[28] = {-1,-1,-1,0,0,0,1,1,1, -1,-1,-1,0,0,0,1,1,1, -1,-1,-1,0,0,0,1,1,1, 0};
constexpr int KDW[28] = {-1,0,1,-1,0,1,-1,0,1, -1,0,1,-1,0,1,-1,0,1, -1,0,1,-1,0,1,-1,0,1, 0};

constexpr int k_off(int k) {  // x-offset of patch element k rel. to (n,0,h,w)
  return KC[k] * 1024 + KDH[k] * 32 + KDW[k];  // multiplies: constexpr-safe
}
constexpr int k_row(int k) {  // weight row start (masked slots -> row 0)
  return (k < KDIM) ? k * OUTF : 0;
}

__device__ __forceinline__ float selu_f(float v) {
  float neg = SELU_ALPHA * expm1f(v);
  return SELU_SCALE * (v > 0.0f ? v : neg);
}

// A-fragment element: KL is the (compile-time) k for lanes 0-15, KH for
// lanes 16-31. Offsets are immediates; validity is a lane-mask combine of
// precomputed predicate bits chosen at compile time from dh/dw. Index is
// unsigned so the load can use the saddr + 32-bit voffset form.
template <int KL, int KH>
__device__ __forceinline__ float a_elem(const float* __restrict__ x,
                                        unsigned xbase, bool lh,
                                        bool hok_m1, bool hok_p1,
                                        bool wok_m1, bool wok_p1) {
  constexpr int dhl = KDH[KL], dwl = KDW[KL];
  constexpr int dhh = KDH[KH], dwh = KDW[KH];
  constexpr unsigned offl = (unsigned)k_off(KL);
  constexpr unsigned offh = (unsigned)k_off(KH);
  constexpr bool klok = (KL < KDIM), khok = (KH < KDIM);

  const bool okl = klok & (dhl == -1 ? hok_m1 : (dhl == 1 ? hok_p1 : true)) &
                          (dwl == -1 ? wok_m1 : (dwl == 1 ? wok_p1 : true));
  const bool okh = khok & (dhh == -1 ? hok_m1 : (dhh == 1 ? hok_p1 : true)) &
                          (dwh == -1 ? wok_m1 : (dwh == 1 ? wok_p1 : true));
  // bitwise combine keeps this in scalar lane-mask ops (no bool materialize)
  const bool ok = (lh & okh) | ((!lh) & okl);
  unsigned idx = xbase + (lh ? offh : offl);
  idx = ok ? idx : 0u;
  float v = x[idx];
  return ok ? v : 0.0f;
}

// One K-step (4 columns) of the GEMM: shared A fragment, 4 feature-group B
// fragments, 4 WMMAs.
template <int T>
__device__ __forceinline__ void k_step(const float* __restrict__ x,
                                       const float* __restrict__ weight,
                                       unsigned xbase, bool lh, unsigned lm,
                                       bool hok_m1, bool hok_p1,
                                       bool wok_m1, bool wok_p1,
                                       v8f (&acc)[4]) {
  constexpr int KL = 4 * T;      // k of element .x, lanes 0-15
  constexpr int KH = 4 * T + 2;  // k of element .x, lanes 16-31

  v2f a;
  a.x = a_elem<KL, KH>(x, xbase, lh, hok_m1, hok_p1, wok_m1, wok_p1);
  a.y = a_elem<KL + 1, KH + 1>(x, xbase, lh, hok_m1, hok_p1, wok_m1, wok_p1);

  // B row offsets: cndmask between two immediates. k>=27 masks are
  // compile-time true except T==6 element .y on the high half.
  const unsigned row0 = lh ? (unsigned)k_row(KH) : (unsigned)k_row(KL);
  const unsigned row1 = lh ? (unsigned)k_row(KH + 1) : (unsigned)k_row(KL + 1);
  constexpr bool kl1ok = (KL + 1 < KDIM), kh1ok = (KH + 1 < KDIM);
  const bool ok1 = (lh & kh1ok) | ((!lh) & kl1ok);  // folds to true for T<6

#pragma unroll
  for (int j = 0; j < 4; ++j) {
    const unsigned f = ((unsigned)j << 4) + lm;
    float b0 = weight[row0 + f];
    float b1 = weight[row1 + f];
    v2f b;
    b.x = b0;                     // .x rows are always < 27
    b.y = ok1 ? b1 : 0.0f;
    acc[j] = __builtin_amdgcn_wmma_f32_16x16x4_f32(
        /*neg_a=*/false, a, /*neg_b=*/false, b,
        /*c_mod=*/(short)0, acc[j], /*reuse_a=*/false, /*reuse_b=*/false);
  }
}

__global__ void zero_ws_kernel(float* __restrict__ p, int count) {
  int i = blockIdx.x * blockDim.x + threadIdx.x;
  if (i < count) p[i] = 0.0f;
}

// One wave = 16 consecutive w-pixels of one (n, h) row x all 64 features.
// GEMM: D[16x64] = A[16x27] * W[27x64] via V_WMMA_F32_16X16X4_F32.
__global__ void __launch_bounds__(256)
conv_selu_pool_kernel(const float* __restrict__ x,
                      const float* __restrict__ weight,
                      const float* __restrict__ bias,
                      float* __restrict__ acc_out) {
  const unsigned wave  = (blockIdx.x << 3) + (threadIdx.x >> 5);
  const unsigned lane  = threadIdx.x & 31;
  const unsigned lm    = lane & 15;
  const bool     lh    = (lane >> 4) != 0;

  const unsigned wtile = wave & 1;       // which half of the 32-wide row
  const unsigned h     = (wave >> 1) & 31;
  const unsigned n     = wave >> 6;
  const unsigned wbase = wtile << 4;
  const unsigned wpix  = wbase + lm;     // pixel column this lane feeds into A

  // base offset of x[n, 0, h, wpix]
  const unsigned xbase = (n * CIN << 10) + (h << 5) + wpix;

  // Halo validity predicates (dh/dw == 0 cases need no check).
  const bool hok_m1 = (h >= 1);
  const bool hok_p1 = (h <= HH - 2);
  const bool wok_m1 = (wpix >= 1);
  const bool wok_p1 = (wpix <= WW - 2);

  v8f acc[4];
  acc[0] = (v8f){};
  acc[1] = (v8f){};
  acc[2] = (v8f){};
  acc[3] = (v8f){};

  k_step<0>(x, weight, xbase, lh, lm, hok_m1, hok_p1, wok_m1, wok_p1, acc);
  k_step<1>(x, weight, xbase, lh, lm, hok_m1, hok_p1, wok_m1, wok_p1, acc);
  k_step<2>(x, weight, xbase, lh, lm, hok_m1, hok_p1, wok_m1, wok_p1, acc);
  k_step<3>(x, weight, xbase, lh, lm, hok_m1, hok_p1, wok_m1, wok_p1, acc);
  k_step<4>(x, weight, xbase, lh, lm, hok_m1, hok_p1, wok_m1, wok_p1, acc);
  k_step<5>(x, weight, xbase, lh, lm, hok_m1, hok_p1, wok_m1, wok_p1, acc);
  k_step<6>(x, weight, xbase, lh, lm, hok_m1, hok_p1, wok_m1, wok_p1, acc);

  // Epilogue: bias + SELU + LPPool(k=2) along w + partial spatial sum.
  // D layout: acc[j] element p -> pixel w = wbase + 8*lh + p, feature
  // f = 16*j + lm. Even/odd pool pairs are adjacent elements of one lane.
#pragma unroll
  for (int j = 0; j < 4; ++j) {
    const unsigned f = ((unsigned)j << 4) + lm;
    const float bf = bias[f];
    v8f d = acc[j];
    float s = 0.0f;
#pragma unroll
    for (int p = 0; p < 4; ++p) {
      float e0 = selu_f(d[2 * p] + bf);
      float e1 = selu_f(d[2 * p + 1] + bf);
      s += sqrtf(e0 * e0 + e1 * e1);
    }
    // combine the two lane-halves (pixels 0-7 and 8-15) -> one atomic per f
    s += __shfl_xor(s, 16, 32);
    if (!lh) {
      atomicAdd(&acc_out[(n << 6) + f], s);
    }
  }
}

// Per-image: channel means of x -> sigmoid gate -> out = (acc/512) * gate.
__global__ void __launch_bounds__(256)
gate_finalize_kernel(const float* __restrict__ x,
                     const float* __restrict__ scale_proj,
                     const float* __restrict__ scale_bias,
                     const float* __restrict__ accbuf,
                     float* __restrict__ out) {
  const int n = blockIdx.x;
  const int t = threadIdx.x;  // 256 threads
  __shared__ float red[3][256];

  const float* xn = x + n * (CIN * HH * WW);
  float s0 = 0.0f, s1 = 0.0f, s2 = 0.0f;
  for (int i = t; i < HH * WW; i += 256) {
    s0 += xn[i];
    s1 += xn[1024 + i];
    s2 += xn[2048 + i];
  }
  red[0][t] = s0;
  red[1][t] = s1;
  red[2][t] = s2;
  __syncthreads();
  for (int off = 128; off > 0; off >>= 1) {
    if (t < off) {
      red[0][t] += red[0][t + off];
      red[1][t] += red[1][t + off];
      red[2][t] += red[2][t + off];
    }
    __syncthreads();
  }
  if (t < OUTF) {
    const float inv_hw = 1.0f / 1024.0f;
    float cs0 = red[0][0] * inv_hw;
    float cs1 = red[1][0] * inv_hw;
    float cs2 = red[2][0] * inv_hw;
    float z = cs0 * scale_proj[t] + cs1 * scale_proj[OUTF + t] +
              cs2 * scale_proj[2 * OUTF + t] + scale_bias[t];
    float gate = 1.0f / (1.0f + expf(-z));
    out[n * OUTF + t] = accbuf[n * OUTF + t] * (1.0f / 512.0f) * gate;
  }
}

extern "C" void kernel_launch(void* const* d_in, const int* in_sizes, int n_in,
                              void* d_out, int out_size, void* d_ws,
                              size_t ws_size, hipStream_t stream) {
  const float* x          = (const float*)d_in[0];  // (1024,3,32,32)
  const float* weight     = (const float*)d_in[1];  // (27,64)
  const float* bias       = (const float*)d_in[2];  // (64,)
  const float* scale_proj = (const float*)d_in[3];  // (3,64)
  const float* scale_bias = (const float*)d_in[4];  // (64,)
  float* out    = (float*)d_out;                    // (1024,64)
  float* accbuf = (float*)d_ws;                     // 1024*64 f32 = 256 KB

  const int acc_elems = NB * OUTF;  // 65536
  zero_ws_kernel<<<(acc_elems + 255) / 256, 256, 0, stream>>>(accbuf,
                                                              acc_elems);

  // waves = 1024 * 32 * 2 = 65536; 8 waves per 256-thread block
  conv_selu_pool_kernel<<<65536 / 8, 256, 0, stream>>>(x, weight, bias,
                                                       accbuf);

  gate_finalize_kernel<<<NB, 256, 0, stream>>>(x, scale_proj, scale_bias,
                                               accbuf, out);
}